// MoshiFlexibleLinearFL_68874095558938
// MI455X (gfx1250) — compile-verified
//
#include <hip/hip_runtime.h>

// ---------------------------------------------------------------------------
// MoshiFlexibleLinear on MI455X (gfx1250, wave32, WMMA, async global->LDS):
//   out[b,s,o] = sum_d x[b,s,d] * weight[layer_idx[s], o, d]
// Per s: GEMM M=32(batch) x N=1024(O) x K=1024(D), fp32 via
// V_WMMA_F32_16X16X4_F32 (exact fp32 semantics; kernel is memory-bound:
// 192 MB min traffic @ 23.3 TB/s ~= 8.2us, weights fit in the 192 MB L2).
// Staging uses GLOBAL_LOAD_ASYNC_TO_LDS_B128 (ASYNCcnt) with double-buffered
// LDS tiles so the async engine fills chunk i+1 while WMMA consumes chunk i.
// ---------------------------------------------------------------------------

typedef __attribute__((ext_vector_type(2))) float v2f;
typedef __attribute__((ext_vector_type(8))) float v8f;

#define NUM_LAYERS 32
#define IN_SIZE    1024
#define OUT_SIZE   1024
#define BATCH      32
#define SEQ        256

#define KC         64          // K chunk staged in LDS
#define NCHUNK     (IN_SIZE / KC)
#define NTILE      128         // N columns per block
#define LDS_STRIDE 66          // 64 + 2 pad: kills ds bank conflicts, keeps 16B rows intact

// Async 16B global->LDS copy (CDNA5 path, tracked by ASYNCcnt).
// LDS operand = low 32 bits of the generic pointer (== LDS byte offset on AMDGPU).
__device__ __forceinline__ void async_copy_b128(const float* gsrc, float* lds_dst) {
    const unsigned ldso = (unsigned)(unsigned long long)lds_dst;
    asm volatile("global_load_async_to_lds_b128 %0, %1, off"
                 :: "v"(ldso), "v"(gsrc)
                 : "memory");
}
__device__ __forceinline__ void wait_async_all() {
    asm volatile("s_wait_asynccnt 0" ::: "memory");
}

__global__ __launch_bounds__(256, 2)
void moshi_flex_linear_wmma(const float* __restrict__ x,
                            const int*   __restrict__ layer_idx,
                            const float* __restrict__ weight,
                            float*       __restrict__ out) {
    // Double-buffered tiles: As[buf][row(batch)][k], Bs[buf][n][k]
    __shared__ float As[2][BATCH * LDS_STRIDE];   // 2 x  8448 B
    __shared__ float Bs[2][NTILE * LDS_STRIDE];   // 2 x 33792 B  (~84.5 KB total)

    const int s  = blockIdx.y;                 // sequence position
    const int n0 = blockIdx.x * NTILE;         // output-column tile base
    const int l  = layer_idx[s];               // expert id (uniform -> scalar load)
    const float* __restrict__ W = weight + (size_t)l * OUT_SIZE * IN_SIZE;

    const int tid  = threadIdx.x;
    const int lane = tid & 31;
    const int wv   = tid >> 5;                 // wave 0..7
    const int mrow = (wv & 1) * 16;            // batch-tile base: 0 | 16
    const int ncol = (wv >> 1) * 32;           // N sub-tile base within block
    const int hi   = lane >> 4;                // lane-half select
    const int lrow = lane & 15;

    v8f acc0 = {};                             // 16x16 tile @ ncol
    v8f acc1 = {};                             // 16x16 tile @ ncol+16

    // Issue all async fills for one K-chunk (10 b128 wave-ops per wave = 40 KB/block).
    auto issue_chunk = [&](int buf, int k0) {
        #pragma unroll
        for (int it = 0; it < 2; ++it) {       // A: 32 rows x 64 floats
            const int idx = tid + it * 256;    // 0..511
            const int row = idx >> 4;
            const int col = (idx & 15) * 4;
            async_copy_b128(x + ((size_t)row * SEQ + s) * IN_SIZE + k0 + col,
                            &As[buf][row * LDS_STRIDE + col]);
        }
        #pragma unroll
        for (int it = 0; it < 8; ++it) {       // B: 128 rows x 64 floats
            const int idx = tid + it * 256;    // 0..2047
            const int n   = idx >> 4;
            const int col = (idx & 15) * 4;
            async_copy_b128(W + (size_t)(n0 + n) * IN_SIZE + k0 + col,
                            &Bs[buf][n * LDS_STRIDE + col]);
        }
    };

    issue_chunk(0, 0);                         // prime the pipeline

    for (int c = 0; c < NCHUNK; ++c) {
        const int buf = c & 1;
        wait_async_all();                      // my chunk-c fills landed
        __syncthreads();                       // everyone's chunk-c fills landed

        if (c + 1 < NCHUNK)                    // overlap: fill other buffer
            issue_chunk(buf ^ 1, (c + 1) * KC);

        const float* Ab = As[buf];
        const float* Bb = Bs[buf];
        #pragma unroll
        for (int kk = 0; kk < KC; kk += 4) {
            // A 16x4 frag: lanes 0-15 K=kk,kk+1 ; lanes 16-31 K=kk+2,kk+3
            const v2f a = *reinterpret_cast<const v2f*>(
                &Ab[(mrow + lrow) * LDS_STRIDE + kk + 2 * hi]);
            // B 4x16 frags (mirrored layout), Bs[n][k] holds W[n][k] = B[k][n]
            const v2f b0 = *reinterpret_cast<const v2f*>(
                &Bb[(ncol + lrow) * LDS_STRIDE + kk + 2 * hi]);
            const v2f b1 = *reinterpret_cast<const v2f*>(
                &Bb[(ncol + 16 + lrow) * LDS_STRIDE + kk + 2 * hi]);
            acc0 = __builtin_amdgcn_wmma_f32_16x16x4_f32(
                false, a, false, b0, (short)0, acc0, false, false);
            acc1 = __builtin_amdgcn_wmma_f32_16x16x4_f32(
                false, a, false, b1, (short)0, acc1, false, false);
        }
        __syncthreads();                       // compute(buf) done before buf is refilled
    }

    // Store: C/D layout — VGPR v holds rows M=v (lanes 0-15) / M=v+8 (lanes 16-31)
    #pragma unroll
    for (int v = 0; v < 8; ++v) {
        const int b  = mrow + v + 8 * hi;
        const int o0 = n0 + ncol + lrow;
        float* dst = out + ((size_t)b * SEQ + s) * OUT_SIZE + o0;
        dst[0]  = acc0[v];
        dst[16] = acc1[v];
    }
}

extern "C" void kernel_launch(void* const* d_in, const int* in_sizes, int n_in,
                              void* d_out, int out_size, void* d_ws, size_t ws_size,
                              hipStream_t stream) {
    const float* x   = (const float*)d_in[0];   // [B, S, D] fp32
    const int*   li  = (const int*)d_in[1];     // [S] int32
    const float* w   = (const float*)d_in[2];   // [L, O, D] fp32
    float*       out = (float*)d_out;           // [B, S, O] fp32

    dim3 grid(OUT_SIZE / NTILE, SEQ);           // (8, 256)
    dim3 block(256);                            // 8 wave32s
    hipLaunchKernelGGL(moshi_flex_linear_wmma, grid, block, 0, stream,
                       x, li, w, out);
}